// SNN_85091892068649
// MI455X (gfx1250) — compile-verified
//
#include <hip/hip_runtime.h>

typedef __attribute__((ext_vector_type(2))) float v2f;
typedef __attribute__((ext_vector_type(8))) float v8f;
typedef int v4i __attribute__((vector_size(16)));  // matches builtin's expected pointee

#define T_STEPS 25
#define BETA 0.9f

#if __has_builtin(__builtin_amdgcn_global_load_async_to_lds_b128) && \
    __has_builtin(__builtin_amdgcn_s_wait_asynccnt)
#define USE_ASYNC_LDS 1
#else
#define USE_ASYNC_LDS 0
#endif

// One wave32 handles 16 batch elements across all 25 timesteps.
// Layer 1 (3->10): one V_WMMA_F32_16X16X4_F32 per t:
//   A = w1 (16x4 f32 A-layout, rows 10..15 / K=3 zero-padded), loop-invariant
//   B = x_t^T (4x16: K=feature, N=batch; lanes 0-15 hold K=0,1, lanes 16-31 hold K=2,pad)
//   C = beta*mem1 + b1 - spk1_prev  (LIF recurrence folded into the WMMA accumulate, in place)
// D layout: VGPR r -> (neuron r, batch=lane) lanes 0-15; (neuron r+8, batch=lane-16) lanes 16-31.
// Layer 2 (10->1): 8 lane-local FMAs + one ds_swizzle SWAPX16 cross-half add.

__global__ __launch_bounds__(256) void SNN_85091892068649_kernel(
    const float* __restrict__ x,   // [B, 25, 3]
    const float* __restrict__ w1,  // [10, 3]
    const float* __restrict__ b1,  // [10]
    const float* __restrict__ w2,  // [1, 10]
    const float* __restrict__ b2,  // [1]
    float* __restrict__ out,       // [B, 25, 1]
    int Btotal)
{
    __shared__ __align__(16) float lds_x[8 * 1200];  // per-wave: 16 rows * 75 floats
    __shared__ __align__(16) float lds_o[8 * 448];   // per-wave: 400 out floats + 48 scratch pad

    const int lane  = threadIdx.x & 31;
    const int wave  = threadIdx.x >> 5;
    const int gwave = blockIdx.x * 8 + wave;
    const long batch_base = (long)gwave * 16;
    if (batch_base >= Btotal) return;  // whole-wave exit only (EXEC all-1 for WMMA)

    float* lx = lds_x + wave * 1200;
    float* lo = lds_o + wave * 448;

    // ---- Stage 16 x-rows (4800 B) into LDS ----
#if USE_ASYNC_LDS
    {
        float* src = const_cast<float*>(x) + batch_base * 75;
        #pragma unroll
        for (int i = 0; i < 9; ++i) {
            const int idx = (lane + i * 32) * 4;
            __builtin_amdgcn_global_load_async_to_lds_b128(
                (__attribute__((address_space(1))) v4i*)(src + idx),
                (__attribute__((address_space(3))) v4i*)(lx + idx), 0, 0);
        }
        if (lane < 12) {
            const int idx = (288 + lane) * 4;
            __builtin_amdgcn_global_load_async_to_lds_b128(
                (__attribute__((address_space(1))) v4i*)(src + idx),
                (__attribute__((address_space(3))) v4i*)(lx + idx), 0, 0);
        }
        __builtin_amdgcn_s_wait_asynccnt(0);
    }
#else
    {
        const float4* src  = (const float4*)(x + batch_base * 75);  // 4800B-aligned
        float4*       dst4 = (float4*)lx;
        #pragma unroll
        for (int i = 0; i < 9; ++i)
            dst4[lane + i * 32] = src[lane + i * 32];
        if (lane < 12)
            dst4[288 + lane] = src[288 + lane];
    }
#endif

    const bool hi = (lane >= 16);
    const int  m  = lane & 15;

    // ---- A operand: w1 in 16x4 f32 A-layout ----
    float a0 = 0.0f, a1 = 0.0f;
    if (m < 10) {
        if (!hi) { a0 = w1[m * 3 + 0]; a1 = w1[m * 3 + 1]; }
        else     { a0 = w1[m * 3 + 2]; /* K=3 pad stays 0 */ }
    }

    // ---- per-half bias / w2 slices aligned with D layout ----
    float b1s[8], w2s[8];
    #pragma unroll
    for (int r = 0; r < 8; ++r) {
        int n = hi ? (r + 8) : r;
        b1s[r] = (n < 10) ? b1[n] : 0.0f;
        w2s[r] = (n < 10) ? w2[n] : 0.0f;
    }
    const float bias2 = b2[0];

    // ---- LIF state ----
    v8f   mem1 = {};
    float spk1[8] = {0.f, 0.f, 0.f, 0.f, 0.f, 0.f, 0.f, 0.f};
    float mem2 = 0.0f, spk2p = 0.0f;

    const float* xrow  = lx + m * 75;
    const int    f0    = hi ? 2 : 0;
    const int    obase = hi ? (400 + m) : (m * 25);  // hi lanes dump duplicates to scratch pad

    #pragma unroll
    for (int t = 0; t < T_STEPS; ++t) {
        // B operand: unconditional LDS loads, mask K=1 slot to 0 on hi lanes (no EXEC dance)
        const float bx = xrow[t * 3 + f0];
        const float by = xrow[t * 3 + 1];
        v2f bop;
        bop.x = bx;
        bop.y = hi ? 0.0f : by;

        // In-place LIF-1 recurrence -> becomes the WMMA C accumulate
        #pragma unroll
        for (int r = 0; r < 8; ++r)
            mem1[r] = __builtin_fmaf(BETA, mem1[r], b1s[r] - spk1[r]);

        v2f aop = {a0, a1};
        mem1 = __builtin_amdgcn_wmma_f32_16x16x4_f32(
            /*neg_a=*/false, aop, /*neg_b=*/false, bop,
            /*c_mod=*/(short)0, mem1, /*reuse_a=*/false, /*reuse_b=*/false);

        // spikes + layer-2 partial dot over this half's neurons
        float partial = 0.0f;
        #pragma unroll
        for (int r = 0; r < 8; ++r) {
            float s = (mem1[r] > 1.0f) ? 1.0f : 0.0f;  // spike == reset predicate
            spk1[r] = s;
            partial = __builtin_fmaf(s, w2s[r], partial);
        }

        // cross-half add: SWAPX16 swizzle (xor=0x10, and=0x1f) -> lane pairs (b, b+16)
        const int   pi    = __builtin_amdgcn_ds_swizzle(__builtin_bit_cast(int, partial), 0x401f);
        const float other = __builtin_bit_cast(float, pi);
        const float cur2  = partial + other + bias2;

        mem2 = __builtin_fmaf(BETA, mem2, cur2 - spk2p);
        const float s2 = (mem2 > 1.0f) ? 1.0f : 0.0f;
        spk2p = s2;

        lo[obase + t] = s2;  // all lanes store; hi-lane duplicates land in scratch pad
    }

    // ---- Coalesced writeback of 16 rows x 25 spikes (1600 B) ----
    {
        const float4* lo4 = (const float4*)lo;
        float4*       o4  = (float4*)(out + batch_base * 25);  // 1600B-aligned
        #pragma unroll
        for (int i = 0; i < 3; ++i)
            o4[lane + i * 32] = lo4[lane + i * 32];
        if (lane < 4)
            o4[96 + lane] = lo4[96 + lane];
    }
}

extern "C" void kernel_launch(void* const* d_in, const int* in_sizes, int n_in,
                              void* d_out, int out_size, void* d_ws, size_t ws_size,
                              hipStream_t stream) {
    const float* x  = (const float*)d_in[0];  // [B, 25, 3]
    const float* w1 = (const float*)d_in[1];  // [10, 3]
    const float* b1 = (const float*)d_in[2];  // [10]
    const float* w2 = (const float*)d_in[3];  // [1, 10]
    const float* b2 = (const float*)d_in[4];  // [1]
    float* out = (float*)d_out;               // [B, 25, 1]

    const int B = in_sizes[0] / (T_STEPS * 3);      // 1048576
    const int waves  = (B + 15) / 16;               // 16 batch per wave
    const int blocks = (waves + 7) / 8;             // 8 waves (256 threads) per block

    SNN_85091892068649_kernel<<<blocks, 256, 0, stream>>>(x, w1, b1, w2, b2, out, B);
}